// DeformConv2dPack_55937654063446
// MI455X (gfx1250) — compile-verified
//
#include <hip/hip_runtime.h>
#include <hip/hip_bf16.h>

#define BB   4
#define CIN  64
#define HH   160
#define WW   160
#define COUT 64
#define KWIN 3
#define KK   9     // KWIN*KWIN
#define CK   576   // CIN*KK

typedef float v2f __attribute__((ext_vector_type(2)));
typedef float v8f __attribute__((ext_vector_type(8)));

// ---------------------------------------------------------------------------
// Kernel 1: offset-producing conv  (B,18,H,W) = conv3x3(x, w_off) + b_off
// ---------------------------------------------------------------------------
__global__ void __launch_bounds__(256)
deform_offset_conv(const float* __restrict__ x,
                   const float* __restrict__ w_off,
                   const float* __restrict__ b_off,
                   float* __restrict__ off) {
  int idx = blockIdx.x * 256 + threadIdx.x;
  if (idx >= BB * 18 * HH * WW) return;
  int w = idx % WW;
  int h = (idx / WW) % HH;
  int o = (idx / (WW * HH)) % 18;
  int b = idx / (WW * HH * 18);

  float acc = b_off[o];
  for (int ky = 0; ky < KWIN; ++ky) {
    int yy = h - 1 + ky;
    if ((unsigned)yy >= HH) continue;
    for (int kx = 0; kx < KWIN; ++kx) {
      int xx = w - 1 + kx;
      if ((unsigned)xx >= WW) continue;
      const float* xp = x + ((size_t)b * CIN * HH + yy) * WW + xx;
      const float* wp = w_off + (size_t)o * CK + ky * KWIN + kx;
      #pragma unroll 8
      for (int c = 0; c < CIN; ++c)
        acc = fmaf(xp[(size_t)c * HH * WW], wp[c * KK], acc);
    }
  }
  off[idx] = acc;
}

// ---------------------------------------------------------------------------
// Kernel 2: fused bilinear-sample + GEMM via V_WMMA_F32_16X16X4_F32.
// Block = 256 threads = 8 wave32 -> 4 M-tiles (16 Cout) x 2 N-tiles (16 px).
// One block produces 64 Cout x 32 pixels; sampled im2col tile lives in LDS.
// ---------------------------------------------------------------------------
#define SW 68   // LDS row stride (floats) for weight slice: 4m mod 64 distinct
#define SS 40   // LDS row stride (floats) for sampled tile: 2*40 mod 64 = 16

__global__ void __launch_bounds__(256, 2)
deform_wmma_gemm(const float* __restrict__ x,
                 const float* __restrict__ weight,
                 const float* __restrict__ bias,
                 const float* __restrict__ off,
                 float* __restrict__ out) {
  __shared__ float s_w[COUT * SW];    // weight[:, :, kk]  (64 x 64, padded)
  __shared__ float s_smp[CIN * SS];   // sampled[c][p]     (64 x 32, padded)
  __shared__ int   s_y0[32], s_x0[32];
  __shared__ float s_ly[32], s_lx[32];

  const int tid   = threadIdx.x;
  const int wtile = blockIdx.x;     // 0..4  (32 pixels along W)
  const int h     = blockIdx.y;     // 0..159
  const int b     = blockIdx.z;     // 0..3

  const int lane  = tid & 31;
  const int wave  = tid >> 5;
  const int mtile = wave >> 1;      // 0..3 : Cout tile
  const int ntile = wave & 1;       // 0..1 : pixel tile
  const int mrow  = lane & 15;      // row (A) / col (B) within 16
  const int kb    = (lane >> 4) << 1;  // lanes 0-15 -> K+0/K+1, 16-31 -> K+2/K+3

  v8f acc = {};
  const float* xb = x + (size_t)b * CIN * HH * WW;

  for (int kk = 0; kk < KK; ++kk) {
    __syncthreads();   // previous iteration's WMMA reads done

    // ---- stage weight slice weight[:, :, kk] into LDS -------------------
    for (int j = tid; j < COUT * CIN; j += 256) {
      int co = j >> 6, c = j & 63;
      s_w[co * SW + c] = weight[(size_t)co * CK + c * KK + kk];
    }

    // ---- per-pixel bilinear metadata (32 pixels) ------------------------
    if (tid < 32) {
      int p  = tid;
      int wp = wtile * 32 + p;
      int ky = kk / 3, kx = kk % 3;
      float dy = off[(((size_t)b * 18 + 2 * kk    ) * HH + h) * WW + wp];
      float dx = off[(((size_t)b * 18 + 2 * kk + 1) * HH + h) * WW + wp];
      float py = (float)(h  - 1 + ky) + dy;
      float px = (float)(wp - 1 + kx) + dx;
      float fy = floorf(py), fx = floorf(px);
      s_y0[p] = (int)fy;  s_x0[p] = (int)fx;
      s_ly[p] = py - fy;  s_lx[p] = px - fx;
    }
    __syncthreads();

    // ---- bilinear gather: 64 channels x 32 pixels, 8 values/thread ------
    {
      int c     = tid >> 2;
      int pbase = (tid & 3) * 8;
      const float* xc = xb + (size_t)c * HH * WW;
      #pragma unroll
      for (int i = 0; i < 8; ++i) {
        int p  = pbase + i;
        int y0 = s_y0[p], x0 = s_x0[p];
        float ly = s_ly[p], lx = s_lx[p];
        int y1 = y0 + 1, x1 = x0 + 1;
        bool yv0 = (unsigned)y0 < HH, yv1 = (unsigned)y1 < HH;
        bool xv0 = (unsigned)x0 < WW, xv1 = (unsigned)x1 < WW;
        float v00 = (yv0 & xv0) ? xc[y0 * WW + x0] : 0.f;
        float v01 = (yv0 & xv1) ? xc[y0 * WW + x1] : 0.f;
        float v10 = (yv1 & xv0) ? xc[y1 * WW + x0] : 0.f;
        float v11 = (yv1 & xv1) ? xc[y1 * WW + x1] : 0.f;
        s_smp[c * SS + p] = v00 * (1.f - ly) * (1.f - lx)
                          + v01 * (1.f - ly) * lx
                          + v10 * ly * (1.f - lx)
                          + v11 * ly * lx;
      }
    }
    __syncthreads();

    // ---- GEMM: 16 K-steps of fp32 WMMA over c = 0..63 -------------------
    const int arow = (mtile * 16 + mrow) * SW;
    const int bcol = ntile * 16 + mrow;
    #pragma unroll
    for (int c0 = 0; c0 < CIN; c0 += 4) {
      v2f a, bf;
      a.x  = s_w[arow + c0 + kb];
      a.y  = s_w[arow + c0 + kb + 1];
      bf.x = s_smp[(c0 + kb)     * SS + bcol];
      bf.y = s_smp[(c0 + kb + 1) * SS + bcol];
      acc = __builtin_amdgcn_wmma_f32_16x16x4_f32(
          /*neg_a=*/false, a, /*neg_b=*/false, bf,
          /*c_mod=*/(short)0, acc, /*reuse_a=*/false, /*reuse_b=*/false);
    }
  }

  // ---- epilogue: bias + store (C/D layout: vgpr r, lane<16 -> M=r else r+8)
  const int nc  = lane & 15;
  const int wp  = wtile * 32 + ntile * 16 + nc;
  const int mhi = (lane >> 4) << 3;
  #pragma unroll
  for (int r = 0; r < 8; ++r) {
    int co = mtile * 16 + mhi + r;
    out[(((size_t)b * COUT + co) * HH + h) * WW + wp] = acc[r] + bias[co];
  }
}

// ---------------------------------------------------------------------------
extern "C" void kernel_launch(void* const* d_in, const int* in_sizes, int n_in,
                              void* d_out, int out_size, void* d_ws, size_t ws_size,
                              hipStream_t stream) {
  const float* x      = (const float*)d_in[0];
  const float* w_off  = (const float*)d_in[1];
  const float* b_off  = (const float*)d_in[2];
  const float* weight = (const float*)d_in[3];
  const float* bias   = (const float*)d_in[4];
  float* out = (float*)d_out;
  float* off = (float*)d_ws;   // 4*18*160*160 floats = 7.37 MB scratch

  (void)in_sizes; (void)n_in; (void)out_size; (void)ws_size;

  int n1 = BB * 18 * HH * WW;
  deform_offset_conv<<<(n1 + 255) / 256, 256, 0, stream>>>(x, w_off, b_off, off);

  dim3 grid(WW / 32, HH, BB);
  deform_wmma_gemm<<<grid, 256, 0, stream>>>(x, weight, bias, off, out);
}